// FLUX_21784074125846
// MI455X (gfx1250) — compile-verified
//
#include <hip/hip_runtime.h>

// Problem constants (from reference setup_inputs)
#define B_CELLS 32768
#define M_MOD   168
#define G_GENES 50
#define H_HID   8
#define C_COMP  70
#define XCOLS   (M_MOD * G_GENES)   // 8400
#define KPAD    52                  // G padded to multiple of 4

typedef __attribute__((ext_vector_type(2))) float v2f;
typedef __attribute__((ext_vector_type(8))) float v8f;

__device__ __forceinline__ float fast_tanhshrink(float x) {
    // tanhshrink(x) = x - tanh(x);   tanh(x) = (e^{2x}-1)/(e^{2x}+1)
    float xc = fminf(fmaxf(x, -15.0f), 15.0f);
    float e  = __expf(2.0f * xc);
    float th = (e - 1.0f) / (e + 1.0f);
    return x - th;
}

// ---------------------------------------------------------------------------
// Kernel 1: per-module tiny MLP.
//   h = tanhshrink(x_m @ W1_m)  [B,8];  m = tanhshrink(h . W2_m + b2_m)
// One block = 1 module x 128 cells (8 waves x 16-cell WMMA tiles).
// WMMA: f32 16x16x4, K looped 0..51 (W1 zero-padded in LDS to [52][16]).
// ---------------------------------------------------------------------------
__global__ void __launch_bounds__(256)
module_mlp_kernel(const float* __restrict__ x,
                  const float* __restrict__ W1,   // [M,50,8]
                  const float* __restrict__ W2,   // [M,8]
                  const float* __restrict__ b2,   // [M]
                  float* __restrict__ out_m)      // [B,M]
{
    __shared__ float w1s[KPAD * 16];  // [K=52][N=16], zero padded

    const int m    = blockIdx.y;
    const int tid  = threadIdx.x;
    const int lane = tid & 31;
    const int wave = tid >> 5;        // 0..7
    const int l15  = lane & 15;
    const int half = (lane < 16) ? 0 : 2;   // K sub-offset for this half-wave

    // Stage W1[m] into LDS, padded (rows 50..51 zero, cols 8..15 zero).
    for (int idx = tid; idx < KPAD * 16; idx += 256) {
        int kk = idx >> 4, hh = idx & 15;
        float v = 0.0f;
        if (kk < G_GENES && hh < H_HID)
            v = W1[((size_t)m * G_GENES + kk) * H_HID + hh];
        w1s[idx] = v;
    }
    __syncthreads();

    const int cell = blockIdx.x * 128 + wave * 16 + l15;    // A-matrix row for this lane
    const float* xrow = x + (size_t)cell * XCOLS + m * G_GENES;

    v8f acc = {};
    #pragma unroll
    for (int k = 0; k < KPAD; k += 4) {
        const int col = k + half;                 // even -> 8B aligned
        v2f a = {0.0f, 0.0f};
        if (col < G_GENES)                        // only lanes>=16 at k=48 masked
            a = *(const v2f*)(xrow + col);
        v2f b;
        b.x = w1s[col * 16 + l15];
        b.y = w1s[(col + 1) * 16 + l15];
        acc = __builtin_amdgcn_wmma_f32_16x16x4_f32(
                  false, a, false, b, (short)0, acc, false, false);
    }

    // tanhshrink on pre-activations (padded cols are 0 -> harmless)
    #pragma unroll
    for (int r = 0; r < 8; ++r) acc[r] = fast_tanhshrink(acc[r]);

    // Second linear: dot over H=8 columns. Lane's column coefficient, then
    // butterfly reduce across the 16-lane half (rows r / r+8).
    const float w2c = (l15 < H_HID) ? W2[(size_t)m * H_HID + l15] : 0.0f;
    float red[8];
    #pragma unroll
    for (int r = 0; r < 8; ++r) {
        float v = acc[r] * w2c;
        v += __shfl_xor(v, 1);
        v += __shfl_xor(v, 2);
        v += __shfl_xor(v, 4);
        v += __shfl_xor(v, 8);
        red[r] = v;   // broadcast within 16-lane half: sum for row r (+8 in hi half)
    }

    if (l15 < 8) {
        float v = red[0];
        #pragma unroll
        for (int r = 1; r < 8; ++r) if (l15 == r) v = red[r];
        const float mv = fast_tanhshrink(v + b2[m]);
        const int rowInTile = l15 + ((lane < 16) ? 0 : 8);
        const int cellOut = blockIdx.x * 128 + wave * 16 + rowInTile;
        out_m[(size_t)cellOut * M_MOD + m] = mv;
    }
}

// ---------------------------------------------------------------------------
// Kernel 2: c[b,i] = sum_mod m[b,mod] * cmMat[i,mod]   ([B,168]x[168,70])
// One block = 16 cells x 80 comps (5 waves, one 16-wide comp tile each).
// K=168 -> exactly 42 WMMA f32 16x16x4 steps; all operands contiguous float2.
// ---------------------------------------------------------------------------
__global__ void __launch_bounds__(160)
compartment_kernel(const float* __restrict__ mvals,  // [B,M]
                   const float* __restrict__ cmMat,  // [C,M]
                   float* __restrict__ out_c)        // [B,C]
{
    const int tid  = threadIdx.x;
    const int lane = tid & 31;
    const int wave = tid >> 5;          // 0..4 -> comp tile
    const int l15  = lane & 15;
    const int half = (lane < 16) ? 0 : 2;

    const int cellBase = blockIdx.x * 16;
    const int comp     = wave * 16 + l15;
    const bool compOK  = (comp < C_COMP);

    const float* arow = mvals + (size_t)(cellBase + l15) * M_MOD;
    const float* brow = cmMat + (size_t)(compOK ? comp : 0) * M_MOD;

    v8f acc = {};
    #pragma unroll 6
    for (int k = 0; k < M_MOD; k += 4) {
        v2f a = *(const v2f*)(arow + k + half);
        v2f b = {0.0f, 0.0f};
        if (compOK) b = *(const v2f*)(brow + k + half);
        acc = __builtin_amdgcn_wmma_f32_16x16x4_f32(
                  false, a, false, b, (short)0, acc, false, false);
    }

    if (compOK) {
        #pragma unroll
        for (int r = 0; r < 8; ++r) {
            const int cell = cellBase + r + ((lane < 16) ? 0 : 8);
            out_c[(size_t)cell * C_COMP + comp] = acc[r];
        }
    }
}

// ---------------------------------------------------------------------------
extern "C" void kernel_launch(void* const* d_in, const int* in_sizes, int n_in,
                              void* d_out, int out_size, void* d_ws, size_t ws_size,
                              hipStream_t stream)
{
    const float* x  = (const float*)d_in[0];   // [B, 8400]
    const float* W1 = (const float*)d_in[1];   // [168,50,8]
    const float* W2 = (const float*)d_in[2];   // [168,8,1]
    const float* b2 = (const float*)d_in[3];   // [168,1]
    const float* cm = (const float*)d_in[4];   // [70,168]
    // d_in[5..7] are the scalar ints n_modules/n_genes/n_comps (compile-time here)

    float* out_m = (float*)d_out;                          // [B, M]
    float* out_c = out_m + (size_t)B_CELLS * M_MOD;        // [B, C]

    dim3 g1(B_CELLS / 128, M_MOD);
    module_mlp_kernel<<<g1, 256, 0, stream>>>(x, W1, W2, b2, out_m);

    compartment_kernel<<<B_CELLS / 16, 160, 0, stream>>>(out_m, cm, out_c);
}